// StableSinkhornKnopp_74955769250029
// MI455X (gfx1250) — compile-verified
//
#include <hip/hip_runtime.h>
#include <hip/hip_bf16.h>

// Sinkhorn-Knopp, separable form: Q[k,b] = r[k] * c[b] * exp((logits[b,k]-M)/EPS).
// Passes: max -> (row-GEMV[WMMA] + finalize -> col-GEMV) x3 -> output. logits (64MB)
// stays L2-resident (192MB L2); exp is recomputed per pass; HBM traffic ~= one
// 64MB read + one 64MB write => ~5.5us floor at 23.3 TB/s.

#define BDIM 16384
#define KDIM 1024
#define INV_EPS 20.0f      // 1/0.05
#define SEPS 1e-9f
#define NCHUNK 32
#define CHUNK_B (BDIM / NCHUNK)   // 512 rows per b-chunk

typedef __attribute__((ext_vector_type(2))) float v2f;
typedef __attribute__((ext_vector_type(8))) float v8f;

#if defined(__gfx1250__) && __has_builtin(__builtin_amdgcn_wmma_f32_16x16x4_f32)
#define HAVE_WMMA_F32 1
#else
#define HAVE_WMMA_F32 0
#endif

__global__ void k_init_c(float* __restrict__ cvec) {
    int i = blockIdx.x * blockDim.x + threadIdx.x;
    if (i < BDIM) cvec[i] = 1.0f;
}

// Stage-1 global max over all 16.7M logits (1024 blocks x 256 threads, float4).
__global__ void k_max1(const float* __restrict__ logits, float* __restrict__ pmax) {
    __shared__ float s[256];
    const int tid = threadIdx.x;
    const int g = blockIdx.x * blockDim.x + tid;
    const int stride = gridDim.x * blockDim.x;
    const float4* L4 = (const float4*)logits;
    const int n4 = (BDIM * KDIM) / 4;
    float m = -3.402823466e+38f;
    for (int i = g; i < n4; i += stride) {
        __builtin_prefetch(&L4[i + stride], 0, 0);   // global_prefetch_b8
        float4 v = L4[i];
        m = fmaxf(m, fmaxf(fmaxf(v.x, v.y), fmaxf(v.z, v.w)));
    }
    s[tid] = m;
    __syncthreads();
    for (int off = 128; off > 0; off >>= 1) {
        if (tid < off) s[tid] = fmaxf(s[tid], s[tid + off]);
        __syncthreads();
    }
    if (tid == 0) pmax[blockIdx.x] = s[0];
}

__global__ void k_max2(const float* __restrict__ pmax, float* __restrict__ Mp) {
    __shared__ float s[256];
    const int tid = threadIdx.x;
    float m = fmaxf(fmaxf(pmax[tid], pmax[tid + 256]),
                    fmaxf(pmax[tid + 512], pmax[tid + 768]));
    s[tid] = m;
    __syncthreads();
    for (int off = 128; off > 0; off >>= 1) {
        if (tid < off) s[tid] = fmaxf(s[tid], s[tid + off]);
        __syncthreads();
    }
    if (tid == 0) Mp[0] = s[0];
}

// Row-reduction GEMV via V_WMMA_F32_16X16X4_F32:
//   rowPart[chunk][k] = sum_{b in chunk} c[b] * E[b,k]
// One wave per (16-wide k-tile, 512-row b-chunk); 2048 waves for latency hiding.
// Per step of 4 rows:
//   A(16x4)[m][kk] = c[b0+kk]  (f32 A layout: lanes0-15 K=0/1 in v0/v1, lanes16-31 K=2/3)
//   B(4x16)[kk][n] = E[b0+kk, kbase+n]  (mirrored striping: v0 = rows 0/2, v1 = rows 1/3)
//   C(16x16)[m][n] = partial rowsum for k=kbase+n, replicated over m.
__global__ void k_rowsum(const float* __restrict__ logits, const float* __restrict__ Mp,
                         const float* __restrict__ cvec, float* __restrict__ rowPart) {
    const int tid = threadIdx.x;
    const int lane = tid & 31;
    const int wave = blockIdx.x * 4 + (tid >> 5);   // 0..2047
    const int ktile = wave & 63;                    // 64 k-tiles
    const int chunk = wave >> 6;                    // 32 b-chunks
    const int kbase = ktile * 16;
    const int n = lane & 15;
    const int half = lane >> 4;
    const int col = kbase + n;
    const float Mv = Mp[0];

#if HAVE_WMMA_F32
    v8f acc = {};
#else
    float acc = 0.0f;
#endif
    const int b_begin = chunk * CHUNK_B;
    const int b_end = b_begin + CHUNK_B;
    for (int b0 = b_begin; b0 < b_end; b0 += 4) {
        const int r0 = b0 + 2 * half;
        const float x0 = logits[r0 * KDIM + col];
        const float x1 = logits[(r0 + 1) * KDIM + col];
        const float e0 = __expf((x0 - Mv) * INV_EPS);
        const float e1 = __expf((x1 - Mv) * INV_EPS);
        const float c0 = cvec[r0];
        const float c1 = cvec[r0 + 1];
#if HAVE_WMMA_F32
        v2f a;  a[0] = c0;  a[1] = c1;
        v2f bm; bm[0] = e0; bm[1] = e1;
        acc = __builtin_amdgcn_wmma_f32_16x16x4_f32(
            /*neg_a=*/false, a, /*neg_b=*/false, bm,
            /*c_mod=*/(short)0, acc, /*reuse_a=*/false, /*reuse_b=*/false);
#else
        float p = c0 * e0 + c1 * e1;     // covers kk={0,1} (lanes<16) / {2,3} (lanes>=16)
        p += __shfl_xor(p, 16, 32);      // combine halves for same column n
        acc += p;
#endif
    }
#if HAVE_WMMA_F32
    const float val = acc[0];            // C[M=0][N=n] on lanes 0..15
#else
    const float val = acc;
#endif
    if (lane < 16) rowPart[chunk * KDIM + kbase + n] = val;
}

// Combine chunk partials in fixed order; update r.
// mode 0: also computes global S = sum(E)+eps, folds the initial Q/=S normalization
// plus iteration-1 row step:  r = (1/S) / ((R/S + eps) * K)
// mode 1: r = r / ((r*R + eps) * K)
__global__ void k_rowfin(const float* __restrict__ rowPart, float* __restrict__ rvec, int mode) {
    __shared__ float sdat[KDIM];
    const int k = threadIdx.x;   // 1024 threads
    float R = 0.0f;
#pragma unroll
    for (int c = 0; c < NCHUNK; ++c) R += rowPart[c * KDIM + k];
    if (mode == 0) {
        sdat[k] = R;
        __syncthreads();
        for (int off = 512; off > 0; off >>= 1) {
            if (k < off) sdat[k] += sdat[k + off];
            __syncthreads();
        }
        const float S = sdat[0] + SEPS;
        const float rowsum = R / S + SEPS;
        rvec[k] = (1.0f / S) / (rowsum * (float)KDIM);
    } else {
        const float rold = rvec[k];
        const float rowsum = rold * R + SEPS;
        rvec[k] = rold / (rowsum * (float)KDIM);
    }
}

// Column step: one wave per sample row b; dot(E[b,:], r) with r staged into LDS via
// CDNA5 async-to-LDS DMA (per-lane global_load_async_to_lds_b128, ASYNCcnt), then
// c[b] = c[b] / ((c[b]*dot + eps) * B).
__global__ void k_colsum(const float* __restrict__ logits, const float* __restrict__ Mp,
                         const float* __restrict__ rvec, float* __restrict__ cvec) {
    __shared__ float sr[KDIM];   // sole LDS allocation -> byte offset 0
    const int tid = threadIdx.x;
#if defined(__gfx1250__)
    {
        // Each lane DMAs 16B: wave w covers sr[ w*128 .. w*128+127 ]; 8 waves = 4KB.
        const float* src = rvec + 4 * tid;
        unsigned lds_off = 16u * (unsigned)tid;     // byte address of sr[4*tid]
        asm volatile("global_load_async_to_lds_b128 %0, %1, off offset:0"
                     :: "v"(lds_off), "v"(src) : "memory");
        asm volatile("s_wait_asynccnt 0" ::: "memory");
    }
#else
    for (int i = tid; i < KDIM; i += 256) sr[i] = rvec[i];
#endif
    __syncthreads();
    const float Mv = Mp[0];
    const int lane = tid & 31;
    const int b = blockIdx.x * 8 + (tid >> 5);
    const float* row = logits + (size_t)b * KDIM;
    float acc = 0.0f;
    for (int k = lane; k < KDIM; k += 32)
        acc = fmaf(__expf((row[k] - Mv) * INV_EPS), sr[k], acc);
#pragma unroll
    for (int off = 16; off > 0; off >>= 1) acc += __shfl_xor(acc, off, 32);
    if (lane == 0) {
        const float cold = cvec[b];
        cvec[b] = cold / ((cold * acc + SEPS) * (float)BDIM);
    }
}

// out[b,k] = B * r[k] * c[b] * E[b,k]   (float4 per thread, k-contiguous)
__global__ void k_out(const float* __restrict__ logits, const float* __restrict__ Mp,
                      const float* __restrict__ rvec, const float* __restrict__ cvec,
                      float* __restrict__ out) {
    const int i4 = blockIdx.x * blockDim.x + threadIdx.x;
    const int base = i4 * 4;
    const int b = base >> 10;
    const int k = base & (KDIM - 1);
    const float Mv = Mp[0];
    const float4 x = ((const float4*)logits)[i4];
    const float4 r = *(const float4*)(rvec + k);
    const float s = (float)BDIM * cvec[b];
    float4 o;
    o.x = s * r.x * __expf((x.x - Mv) * INV_EPS);
    o.y = s * r.y * __expf((x.y - Mv) * INV_EPS);
    o.z = s * r.z * __expf((x.z - Mv) * INV_EPS);
    o.w = s * r.w * __expf((x.w - Mv) * INV_EPS);
    ((float4*)out)[i4] = o;
}

extern "C" void kernel_launch(void* const* d_in, const int* in_sizes, int n_in,
                              void* d_out, int out_size, void* d_ws, size_t ws_size,
                              hipStream_t stream) {
    (void)in_sizes; (void)n_in; (void)out_size; (void)ws_size;
    const float* logits = (const float*)d_in[0];
    float* out = (float*)d_out;

    // Scratch layout (floats), ~204 KB total, fully overwritten every call:
    float* w       = (float*)d_ws;
    float* pmax    = w;            // [1024]    stage-1 max partials
    float* Mp      = w + 1024;     // [1]       global max
    float* rowPart = w + 2048;     // [32*1024] row-sum chunk partials
    float* rvec    = w + 34816;    // [1024]    prototype scales r[k]
    float* cvec    = w + 35840;    // [16384]   sample scales c[b]

    k_init_c<<<BDIM / 256, 256, 0, stream>>>(cvec);
    k_max1<<<1024, 256, 0, stream>>>(logits, pmax);
    k_max2<<<1, 256, 0, stream>>>(pmax, Mp);

    // Iteration 1 (row step folds the global-sum normalization), then 2 more.
    k_rowsum<<<512, 128, 0, stream>>>(logits, Mp, cvec, rowPart);
    k_rowfin<<<1, KDIM, 0, stream>>>(rowPart, rvec, 0);
    k_colsum<<<BDIM / 8, 256, 0, stream>>>(logits, Mp, rvec, cvec);
    for (int it = 1; it < 3; ++it) {
        k_rowsum<<<512, 128, 0, stream>>>(logits, Mp, cvec, rowPart);
        k_rowfin<<<1, KDIM, 0, stream>>>(rowPart, rvec, 1);
        k_colsum<<<BDIM / 8, 256, 0, stream>>>(logits, Mp, rvec, cvec);
    }

    k_out<<<(BDIM * KDIM) / 1024, 256, 0, stream>>>(logits, Mp, rvec, cvec, out);
}